// SupErosionsBlock_62861141344687
// MI455X (gfx1250) — compile-verified
//
#include <hip/hip_runtime.h>
#include <hip/hip_bf16.h>
#include <stdint.h>

// Problem constants (from reference): x (16,128,128,3) f32, w (3,3,3,16) f32,
// out (16,126,126,1) f32. Erosion via min-plus, then max over 16 erosions.
constexpr int N_  = 16;
constexpr int H_  = 128;
constexpr int W_  = 128;
constexpr int C_  = 3;
constexpr int F_  = 16;
constexpr int KH_ = 3;
constexpr int KW_ = 3;
constexpr int OH_ = H_ - KH_ + 1;   // 126
constexpr int OW_ = W_ - KW_ + 1;   // 126
constexpr int NTAP = KH_ * KW_ * C_; // 27

// Tiling: one 256-thread block (8 wave32) -> 32 (wo) x 8 (ho) outputs.
constexpr int TW_    = 32;
constexpr int THh_   = 8;
constexpr int TROWS  = THh_ + KH_ - 1;       // 10 input rows
constexpr int TCOLS  = TW_  + KW_ - 1;       // 34 input cols
constexpr int ROWF   = TCOLS * C_;           // 102 floats per tile row
constexpr int TILEN  = TROWS * ROWF;         // 1020 floats (4080 B LDS)

// Encourage v_min3_num_f32 / v_max3_num_f32 folding.
__device__ __forceinline__ float min3f(float a, float b, float c) {
    return fminf(fminf(a, b), c);
}
__device__ __forceinline__ float max3f(float a, float b, float c) {
    return fmaxf(fmaxf(a, b), c);
}

__global__ __launch_bounds__(256, 4)
void sup_erosion_kernel(const float* __restrict__ x,
                        const float* __restrict__ w,
                        float* __restrict__ out) {
    __shared__ float tile[TILEN];

    const int tid = threadIdx.x;
    const int wo0 = blockIdx.x * TW_;
    const int ho0 = blockIdx.y * THh_;
    const int n   = blockIdx.z;

    const float* xn = x + (size_t)n * H_ * W_ * C_;

    // ---- Stage input halo tile into LDS with CDNA5 async global->LDS DMA ----
    // Division-free distribution: 5 rounds x 204 active lanes (2 rows of 102
    // contiguous floats per round). A tile row is contiguous in memory, so
    // right-edge clamping is a single min() on the flat in-row element index;
    // bottom-edge clamping is a min() on the row. Clamped (duplicate) values
    // feed only discarded outputs.
    {
        const int  p    = (tid >= ROWF) ? 1 : 0;       // row parity in pair
        const int  e    = tid - p * ROWF;              // 0..101 in-row element
        const int  emax = (W_ - 1 - wo0) * C_ + (C_ - 1); // last valid element
        const int  ec   = (e < emax) ? e : emax;       // clamp right edge
        const int  lbase = p * ROWF + e;               // LDS element index
        const int  gcol  = wo0 * C_ + ec;              // in-row float offset
        if (tid < 2 * ROWF) {
            #pragma unroll
            for (int k = 0; k < 5; ++k) {
                int r  = 2 * k + p;                    // tile row 0..9
                int gr = ho0 + r; if (gr > H_ - 1) gr = H_ - 1;
                const float* src = xn + (size_t)gr * (W_ * C_) + gcol;
                uint32_t lds_off =
                    (uint32_t)(uintptr_t)(&tile[k * 2 * ROWF + lbase]);
                asm volatile("global_load_async_to_lds_b32 %0, %1, off"
                             :: "v"(lds_off), "v"(src)
                             : "memory");
            }
        }
    }
    asm volatile("s_wait_asynccnt 0" ::: "memory");
    __syncthreads();

    // ---- Preload all 27 tap values into registers (LDS reads off one base) ----
    const int tc = tid & (TW_ - 1);   // local wo
    const int tr = tid >> 5;          // local ho

    float xv[NTAP];
    {
        const int base0 = tr * ROWF + tc * C_;
        #pragma unroll
        for (int i = 0; i < KH_; ++i) {
            #pragma unroll
            for (int j = 0; j < KW_; ++j) {
                int base = base0 + i * ROWF + j * C_;  // constant offsets
                #pragma unroll
                for (int c = 0; c < C_; ++c) {
                    xv[(i * KW_ + j) * C_ + c] = tile[base + c];
                }
            }
        }
    }

    // ---- Per filter: 27 v_sub_f32 (SGPR weight operands) + min3 tree ----
    // 27 -> 9 -> 3 -> 1 via v_min3_num_f32 (13 instrs/filter vs 26 v_min).
    float ero[F_];
    #pragma unroll
    for (int f = 0; f < F_; ++f) {
        float d[NTAP];
        #pragma unroll
        for (int t = 0; t < NTAP; ++t) {
            // w layout (kh,kw,C,F): tap t, filter f -> w[t*F_ + f]; uniform
            // index -> scalar s_load, SGPR src0 of v_sub_f32.
            d[t] = xv[t] - w[t * F_ + f];
        }
        float g[9];
        #pragma unroll
        for (int q = 0; q < 9; ++q) {
            g[q] = min3f(d[3 * q], d[3 * q + 1], d[3 * q + 2]);
        }
        float h0 = min3f(g[0], g[1], g[2]);
        float h1 = min3f(g[3], g[4], g[5]);
        float h2 = min3f(g[6], g[7], g[8]);
        ero[f] = min3f(h0, h1, h2);
    }

    // ---- 16-way max via max3 tree: 16 -> 6 -> 2 -> 1 ----
    float m0 = max3f(ero[0],  ero[1],  ero[2]);
    float m1 = max3f(ero[3],  ero[4],  ero[5]);
    float m2 = max3f(ero[6],  ero[7],  ero[8]);
    float m3 = max3f(ero[9],  ero[10], ero[11]);
    float m4 = max3f(ero[12], ero[13], ero[14]);
    float m5 = ero[15];
    float p0 = max3f(m0, m1, m2);
    float p1 = max3f(m3, m4, m5);
    float m  = fmaxf(p0, p1);

    const int ho = ho0 + tr;
    const int wo = wo0 + tc;
    if (ho < OH_ && wo < OW_) {
        out[((size_t)n * OH_ + ho) * OW_ + wo] = m;
    }
}

extern "C" void kernel_launch(void* const* d_in, const int* in_sizes, int n_in,
                              void* d_out, int out_size, void* d_ws, size_t ws_size,
                              hipStream_t stream) {
    (void)in_sizes; (void)n_in; (void)out_size; (void)d_ws; (void)ws_size;
    const float* x = (const float*)d_in[0];
    const float* w = (const float*)d_in[1];
    float* out     = (float*)d_out;

    dim3 grid((OW_ + TW_ - 1) / TW_,    // 4
              (OH_ + THh_ - 1) / THh_,  // 16
              N_);                      // 16
    dim3 block(256);
    hipLaunchKernelGGL(sup_erosion_kernel, grid, block, 0, stream, x, w, out);
}